// GCNLayer_29557964931202
// MI455X (gfx1250) — compile-verified
//
#include <hip/hip_runtime.h>
#include <hip/hip_bf16.h>
#include <math.h>

// GCN layer for MI455X (gfx1250):
//   out = D^-1/2 (A + I) D^-1/2 (X W) + b
// N = 8192, D_IN = D_OUT = 128, all fp32.
//
// Pipeline (all on `stream`):
//   1. zero_kernel      : d_out = 0                         (needed for split-K atomics)
//   2. degree_kernel    : dinv[i] = rsqrt(1 + rowsum(A,i))  (one 256MB pass over A)
//   3. support_kernel   : s2 = dinv[:,None] * (X @ W)       (tiny GEMM, VALU)
//   4. spmm_kernel      : d_out += A @ s2 (WMMA f32 16x16x4, split-K=16,
//                         double-buffered async global->LDS tiles)
//   5. finalize_kernel  : d_out = dinv[i]*(d_out + s2) + bias

#define GCN_N     8192
#define GCN_D     128
#define BM        128
#define BK        32
#define LDA       (BK + 4)      // 36 floats: 16B-aligned b128 dsts, conflict-free A frags
#define LDB       (GCN_D + 4)   // 132 floats: 16B-aligned b128 dsts for B rows
#define SBUF      (BM * LDA + BK * LDB)   // floats per LDS buffer (4608 + 4224 = 8832)
#define KCHUNKS   16
#define KCHUNK    (GCN_N / KCHUNKS)       // 512

typedef float v2f __attribute__((ext_vector_type(2)));
typedef float v8f __attribute__((ext_vector_type(8)));
typedef int   v4i __attribute__((ext_vector_type(4)));

// Pointer flavors for the async global->LDS DMA builtin (per hipcc diagnostic:
// param 0 is 'int __vector(4) __device__ *', i.e. v4i in addrspace(1)).
typedef v4i __attribute__((address_space(1))) gas_v4i;   // global
typedef v4i __attribute__((address_space(3))) las_v4i;   // LDS

// ---- CDNA5 async global->LDS path (fire-and-forget DMA tracked by ASYNCcnt)
#if defined(__has_builtin)
#if __has_builtin(__builtin_amdgcn_global_load_async_to_lds_b128) && \
    __has_builtin(__builtin_amdgcn_s_wait_asynccnt)
#define GCN_HAVE_ASYNC 1
#endif
#endif
#ifndef GCN_HAVE_ASYNC
#define GCN_HAVE_ASYNC 0
#endif

__device__ __forceinline__ void gcn_copy_b128(const float* __restrict__ g,
                                              float* __restrict__ l) {
#if GCN_HAVE_ASYNC
    __builtin_amdgcn_global_load_async_to_lds_b128(
        (gas_v4i*)(g),            // global source (AS1, const cast away)
        (las_v4i*)(l),            // LDS destination (AS3)
        /*offset=*/0, /*cpol=*/0);
#else
    *(float4*)l = *(const float4*)g;
#endif
}

// ---------------------------------------------------------------- zero d_out
__global__ void gcn_zero_kernel(float4* __restrict__ out) {
    out[blockIdx.x * blockDim.x + threadIdx.x] = make_float4(0.f, 0.f, 0.f, 0.f);
}

// ------------------------------------------------- dinv[i] = rsqrt(1+rowsum)
__global__ void gcn_degree_kernel(const float* __restrict__ A,
                                  float* __restrict__ dinv) {
    __shared__ float red[256];
    const int row = blockIdx.x;
    const float4* Arow = (const float4*)(A + (size_t)row * GCN_N);
    float s = 0.f;
    for (int i = threadIdx.x; i < GCN_N / 4; i += 256) {
        float4 v = Arow[i];
        s += v.x + v.y + v.z + v.w;
    }
    red[threadIdx.x] = s;
    __syncthreads();
    for (int off = 128; off > 0; off >>= 1) {
        if (threadIdx.x < off) red[threadIdx.x] += red[threadIdx.x + off];
        __syncthreads();
    }
    if (threadIdx.x == 0) {
        float d = red[0] + 1.0f;             // A_hat = A + I
        dinv[row] = (d > 0.f) ? rsqrtf(d) : 0.f;
    }
}

// ----------------------------------------- s2 = dinv[:,None] * (X @ W) [N,D]
__global__ __launch_bounds__(128)
void gcn_support_kernel(const float* __restrict__ X,
                        const float* __restrict__ W,
                        const float* __restrict__ dinv,
                        float* __restrict__ s2) {
    __shared__ float xs[8][GCN_D];
    const int c = threadIdx.x;
    const int rowBase = blockIdx.x * 8;
    #pragma unroll
    for (int r = 0; r < 8; ++r)
        xs[r][c] = X[(size_t)(rowBase + r) * GCN_D + c];
    __syncthreads();
    float acc[8] = {0.f, 0.f, 0.f, 0.f, 0.f, 0.f, 0.f, 0.f};
    for (int k = 0; k < GCN_D; ++k) {
        float w = W[(size_t)k * GCN_D + c];
        #pragma unroll
        for (int r = 0; r < 8; ++r) acc[r] += xs[r][k] * w;
    }
    #pragma unroll
    for (int r = 0; r < 8; ++r)
        s2[(size_t)(rowBase + r) * GCN_D + c] = acc[r] * dinv[rowBase + r];
}

// -------------------------------------------- cooperative tile loader
// A tile: BM x BK into As[r*LDA + k]; B tile: BK x 128 row-major into Bs[k*LDB + c]
__device__ __forceinline__ void gcn_load_tiles(const float* __restrict__ A,
                                               const float* __restrict__ B,
                                               float* __restrict__ As,
                                               float* __restrict__ Bs,
                                               int rowBlock, int kc, int tid) {
    #pragma unroll
    for (int i = 0; i < 4; ++i) {
        int f4 = tid + i * 256;            // 0..1023 over 128x8 float4s
        int r  = f4 >> 3;
        int c4 = f4 & 7;
        gcn_copy_b128(A + (size_t)(rowBlock + r) * GCN_N + kc + c4 * 4,
                      &As[r * LDA + c4 * 4]);
    }
    #pragma unroll
    for (int i = 0; i < 4; ++i) {
        int f4 = tid + i * 256;            // 0..1023 over 32x32 float4s
        int r  = f4 >> 5;                  // k-row 0..31
        int c4 = f4 & 31;
        gcn_copy_b128(B + (size_t)(kc + r) * GCN_D + c4 * 4,
                      &Bs[r * LDB + c4 * 4]);
    }
}

// -------------------------------------------- d_out += A @ s2  (WMMA, splitK)
// grid = (64 row-blocks, 16 K-chunks), block = 256 threads = 8 waves.
// Double-buffered LDS ping-pong with async global->LDS DMA.
__global__ __launch_bounds__(256)
void gcn_spmm_kernel(const float* __restrict__ A,
                     const float* __restrict__ B,      // s2 [N, 128]
                     float* __restrict__ out) {
    extern __shared__ float lds[];

    const int tid  = threadIdx.x;
    const int wave = tid >> 5;
    const int lane = tid & 31;
    const int half = lane >> 4;           // 0: lanes 0-15, 1: lanes 16-31
    const int m16  = lane & 15;
    const int wm   = wave >> 1;           // 0..3 (M direction)
    const int wn   = wave & 1;            // 0..1 (N direction)
    const int rowBlock = blockIdx.x * BM;
    const int kStart   = blockIdx.y * KCHUNK;

    v8f acc[2][4];
    #pragma unroll
    for (int mt = 0; mt < 2; ++mt)
        #pragma unroll
        for (int nt = 0; nt < 4; ++nt) acc[mt][nt] = (v8f){};

    gcn_load_tiles(A, B, lds, lds + BM * LDA, rowBlock, kStart, tid);

    const int nTiles = KCHUNK / BK;       // 16
    for (int t = 0; t < nTiles; ++t) {
        const int curOff  = (t & 1) ? SBUF : 0;   // integer buffer select:
        const int nextOff = (t & 1) ? 0 : SBUF;   // no LDS pointer arrays
#if GCN_HAVE_ASYNC
        __builtin_amdgcn_s_wait_asynccnt(0);
#endif
        __syncthreads();                  // current buffer fully resident

        if (t + 1 < nTiles)               // kick off next tile's DMA
            gcn_load_tiles(A, B, lds + nextOff, lds + nextOff + BM * LDA,
                           rowBlock, kStart + (t + 1) * BK, tid);

        const float* As = lds + curOff;
        const float* Bs = lds + curOff + BM * LDA;

        #pragma unroll
        for (int k0 = 0; k0 < BK; k0 += 4) {
            v2f a[2], b[4];
            // A frag (16x4 f32): lanes 0-15 -> K=k0,k0+1 ; lanes 16-31 -> k0+2,k0+3
            #pragma unroll
            for (int mt = 0; mt < 2; ++mt) {
                int row = wm * 32 + mt * 16 + m16;
                a[mt] = *(const v2f*)(&As[row * LDA + k0 + half * 2]);
            }
            // B frag (4x16 f32) from row-major LDS: two b32 per lane
            #pragma unroll
            for (int nt = 0; nt < 4; ++nt) {
                int col = wn * 64 + nt * 16 + m16;
                int kk  = k0 + half * 2;
                b[nt] = (v2f){ Bs[kk * LDB + col], Bs[(kk + 1) * LDB + col] };
            }
            #pragma unroll
            for (int mt = 0; mt < 2; ++mt)
                #pragma unroll
                for (int nt = 0; nt < 4; ++nt)
                    acc[mt][nt] = __builtin_amdgcn_wmma_f32_16x16x4_f32(
                        /*neg_a=*/false, a[mt], /*neg_b=*/false, b[nt],
                        /*c_mod=*/(short)0, acc[mt][nt],
                        /*reuse_a=*/false, /*reuse_b=*/false);
        }
        __syncthreads();                  // done reading cur before it's reloaded
    }

    // ---- split-K accumulate into d_out (L2-resident fp32 atomics)
    // C/D layout: VGPR v holds M = v + 8*half, N = m16 (within its 16x16 tile)
    #pragma unroll
    for (int mt = 0; mt < 2; ++mt) {
        #pragma unroll
        for (int nt = 0; nt < 4; ++nt) {
            int col = wn * 64 + nt * 16 + m16;
            #pragma unroll
            for (int v = 0; v < 8; ++v) {
                int row = rowBlock + wm * 32 + mt * 16 + v + half * 8;
                atomicAdd(&out[(size_t)row * GCN_D + col], acc[mt][nt][v]);
            }
        }
    }
}

// ------------------------- out = dinv[i]*(out + s2) + bias  (identity + norm)
__global__ void gcn_finalize_kernel(float* __restrict__ out,
                                    const float* __restrict__ s2,
                                    const float* __restrict__ dinv,
                                    const float* __restrict__ bias) {
    int idx = blockIdx.x * blockDim.x + threadIdx.x;   // < 8192*128
    int row = idx >> 7;
    int col = idx & (GCN_D - 1);
    out[idx] = dinv[row] * (out[idx] + s2[idx]) + bias[col];
}

extern "C" void kernel_launch(void* const* d_in, const int* in_sizes, int n_in,
                              void* d_out, int out_size, void* d_ws, size_t ws_size,
                              hipStream_t stream) {
    const float* X    = (const float*)d_in[0];   // node_features [8192,128]
    const float* Adj  = (const float*)d_in[1];   // adj_matrix    [8192,8192]
    const float* W    = (const float*)d_in[2];   // weight        [128,128]
    const float* bias = (const float*)d_in[3];   // bias          [128]
    float* out = (float*)d_out;                  // [8192,128]

    float* ws   = (float*)d_ws;
    float* dinv = ws;                            // 8192 floats
    float* s2   = ws + 8192;                     // 8192*128 floats (~4MB)

    gcn_zero_kernel<<<(GCN_N * GCN_D / 4) / 256, 256, 0, stream>>>((float4*)out);
    gcn_degree_kernel<<<GCN_N, 256, 0, stream>>>(Adj, dinv);
    gcn_support_kernel<<<GCN_N / 8, 128, 0, stream>>>(X, W, dinv, s2);

    dim3 grid(GCN_N / BM, KCHUNKS);
    size_t ldsBytes = (size_t)(2 * SBUF) * sizeof(float);   // ~69 KB (gfx1250: 320KB/WG)
    gcn_spmm_kernel<<<grid, 256, ldsBytes, stream>>>(Adj, s2, out);

    gcn_finalize_kernel<<<(GCN_N * GCN_D) / 256, 256, 0, stream>>>(out, s2, dinv, bias);
}